// GCN_node_sparse_56178172232071
// MI455X (gfx1250) — compile-verified
//
#include <hip/hip_runtime.h>

typedef float v2f __attribute__((ext_vector_type(2)));
typedef float v8f __attribute__((ext_vector_type(8)));

#define GCN_N_NODES 100000
#define GCN_N_EDGES 1600000
#define GCN_F_IN    128
#define GCN_F_HID   128
#define GCN_F_OUT   40

// ---------------- degree / normalization ----------------

__global__ void gcn_deg_init(float* __restrict__ deg, int n) {
    int i = blockIdx.x * blockDim.x + threadIdx.x;
    if (i < n) deg[i] = 1.0f;  // self-loop weight
}

__global__ void gcn_deg_accum(const long long* __restrict__ dst,
                              const float* __restrict__ w,
                              float* __restrict__ deg, int e) {
    int i = blockIdx.x * blockDim.x + threadIdx.x;
    if (i < e) atomicAdd(deg + (int)dst[i], w[i]);
}

__global__ void gcn_deg_to_dinv(float* __restrict__ deg, int n) {
    int i = blockIdx.x * blockDim.x + threadIdx.x;
    if (i < n) {
        float d = deg[i];
        deg[i] = (d > 0.0f) ? rsqrtf(d) : 0.0f;
    }
}

// ---------------- fp32 WMMA GEMM:  C[M,N] = op(A)[M,K] @ B[K,N] ----------------
// One wave computes one 16x16 output tile via 16x16x4 f32 WMMA, K/4 steps.
// RELU != 0 applies max(0,.) to A elements on load (fuses previous layer's ReLU).
// M must be a multiple of 16 (100000 = 6250*16). K, N are compile-time.
// When N % 16 == 0 the column guard is compile-time dead: B loads feed WMMA
// directly. Ragged N (40) uses a clamped column + select-to-zero (branch-free).

template <int RELU, int K, int N>
__global__ void gcn_wmma_gemm(const float* __restrict__ A,
                              const float* __restrict__ B,
                              float* __restrict__ C,
                              int tilesM, int tilesN) {
    constexpr bool RAGGED = (N % 16) != 0;

    const int lane  = threadIdx.x & 31;
    const int wave  = threadIdx.x >> 5;
    const int tile  = blockIdx.x * (blockDim.x >> 5) + wave;
    if (tile >= tilesM * tilesN) return;
    const int tm = tile / tilesN;
    const int tn = tile % tilesN;

    const int r16   = lane & 15;          // A row within tile / B,C column within tile
    const int khalf = (lane >> 4) << 1;   // 0 for lanes 0-15, 2 for lanes 16-31

    const int  ncol_raw = tn * 16 + r16;
    const bool nok      = RAGGED ? (ncol_raw < N) : true;
    const int  ncol     = RAGGED ? (nok ? ncol_raw : (N - 1)) : ncol_raw;
    const float bmask   = (RAGGED && !nok) ? 0.0f : 1.0f;

    const float* __restrict__ arow  = A + (size_t)(tm * 16 + r16) * K + khalf;
    const float* __restrict__ bbase = B + (size_t)khalf * N + ncol;

    v8f acc = {};
#pragma unroll
    for (int k0 = 0; k0 < K; k0 += 4) {
        // A fragment (16x4 f32): lane holds A[r16][k0+khalf .. +1]  (contiguous, b64 load)
        float a0 = arow[k0];
        float a1 = arow[k0 + 1];
        if (RELU) { a0 = fmaxf(a0, 0.0f); a1 = fmaxf(a1, 0.0f); }
        v2f a; a.x = a0; a.y = a1;

        // B fragment (4x16 f32): lane holds B[k0+khalf .. +1][ncol] (immediate offsets)
        v2f b;
        if (RAGGED) {
            b.x = bbase[(size_t)k0 * N]       * bmask;
            b.y = bbase[(size_t)(k0 + 1) * N] * bmask;
        } else {
            b.x = bbase[(size_t)k0 * N];
            b.y = bbase[(size_t)(k0 + 1) * N];
        }

        acc = __builtin_amdgcn_wmma_f32_16x16x4_f32(
            /*neg_a=*/false, a, /*neg_b=*/false, b,
            /*c_mod=*/(short)0, acc, /*reuse_a=*/false, /*reuse_b=*/false);
    }

    // C/D layout: VGPR r -> row (r + 8*(lane>=16)), column = lane&15
    if (nok) {
        float* __restrict__ cptr = C + (size_t)(tm * 16 + ((lane >> 4) << 3)) * N + ncol_raw;
#pragma unroll
        for (int r = 0; r < 8; ++r)
            cptr[(size_t)r * N] = acc[r];
    }
}

// ---------------- aggregation ----------------
// out[v][f] = bias[f] + h[v][f] * dinv[v]^2   (self-loop term seeds the buffer)
__global__ void gcn_agg_init(const float* __restrict__ h,
                             const float* __restrict__ dinv,
                             const float* __restrict__ bias,
                             float* __restrict__ out,
                             int n, int F) {
    int i = blockIdx.x * blockDim.x + threadIdx.x;
    if (i >= n * F) return;
    int v = i / F, f = i - v * F;
    float di = dinv[v];
    out[i] = bias[f] + h[i] * di * di;
}

// One wave per edge; lanes stripe the feature dimension.
// out[dst][f] += h[src][f] * dinv[src]*w*dinv[dst]
template <int F>
__global__ void gcn_edge_scatter(const long long* __restrict__ src,
                                 const long long* __restrict__ dst,
                                 const float* __restrict__ w,
                                 const float* __restrict__ dinv,
                                 const float* __restrict__ h,
                                 float* __restrict__ out,
                                 int e) {
    int lane = threadIdx.x & 31;
    int edge = (blockIdx.x * blockDim.x + threadIdx.x) >> 5;
    if (edge >= e) return;
    int s = (int)src[edge];
    int d = (int)dst[edge];
    float norm = dinv[s] * w[edge] * dinv[d];
    const float* __restrict__ hs = h + (size_t)s * F;
    float*       __restrict__ od = out + (size_t)d * F;

    if (F == 128) {
        // one coalesced 512B row per wave: float4 (b128) gather, 4 atomics per lane
        const float4 hv = *(const float4*)(hs + lane * 4);
        atomicAdd(od + lane * 4 + 0, hv.x * norm);
        atomicAdd(od + lane * 4 + 1, hv.y * norm);
        atomicAdd(od + lane * 4 + 2, hv.z * norm);
        atomicAdd(od + lane * 4 + 3, hv.w * norm);
    } else {
#pragma unroll
        for (int f = lane; f < F; f += 32)
            atomicAdd(od + f, hs[f] * norm);
    }
}

// ---------------- orchestration ----------------

extern "C" void kernel_launch(void* const* d_in, const int* in_sizes, int n_in,
                              void* d_out, int out_size, void* d_ws, size_t ws_size,
                              hipStream_t stream) {
    const float*     x   = (const float*)d_in[0];
    const long long* ei  = (const long long*)d_in[1];   // jnp.int64, [2, E]
    const float*     ew  = (const float*)d_in[2];
    const float*     W1  = (const float*)d_in[3];
    const float*     b1  = (const float*)d_in[4];
    const float*     W2  = (const float*)d_in[5];
    const float*     b2  = (const float*)d_in[6];
    const float*     W3  = (const float*)d_in[7];
    const float*     b3  = (const float*)d_in[8];
    float*           out = (float*)d_out;

    const long long* src = ei;
    const long long* dst = ei + GCN_N_EDGES;

    // workspace carving (256B aligned)
    char* ws = (char*)d_ws;
    float* dinv  = (float*)(ws);                                  // 400 KB
    float* bufH  = (float*)(ws + ((size_t)409600));               // 51.2 MB  (N x 128)
    float* bufA  = (float*)(ws + ((size_t)409600 + 51200000));    // 51.2 MB  (N x 128)
    float* bufH3 = (float*)(ws + ((size_t)409600 + 2*51200000));  // 16 MB    (N x 40)

    const int TB = 256;
    const int nBlkNodes = (GCN_N_NODES + TB - 1) / TB;
    const int nBlkEdges = (GCN_N_EDGES + TB - 1) / TB;
    const int nBlkEdgeWaves = (GCN_N_EDGES * 32 + TB - 1) / TB;  // wave per edge
    const int tilesM = GCN_N_NODES / 16;                          // 6250, exact

    // ---- normalization: deg -> dinv ----
    gcn_deg_init<<<nBlkNodes, TB, 0, stream>>>(dinv, GCN_N_NODES);
    gcn_deg_accum<<<nBlkEdges, TB, 0, stream>>>(dst, ew, dinv, GCN_N_EDGES);
    gcn_deg_to_dinv<<<nBlkNodes, TB, 0, stream>>>(dinv, GCN_N_NODES);

    // ---- layer 1: h = x @ W1 ; agg -> bufA ----
    {
        const int tilesN = GCN_F_HID / 16;  // 8
        const int blocks = (tilesM * tilesN + (TB / 32) - 1) / (TB / 32);
        gcn_wmma_gemm<0, GCN_F_IN, GCN_F_HID><<<blocks, TB, 0, stream>>>(
            x, W1, bufH, tilesM, tilesN);
        const int elems = GCN_N_NODES * GCN_F_HID;
        gcn_agg_init<<<(elems + TB - 1) / TB, TB, 0, stream>>>(
            bufH, dinv, b1, bufA, GCN_N_NODES, GCN_F_HID);
        gcn_edge_scatter<GCN_F_HID><<<nBlkEdgeWaves, TB, 0, stream>>>(
            src, dst, ew, dinv, bufH, bufA, GCN_N_EDGES);
    }

    // ---- layer 2: h = relu(agg1) @ W2 ; agg -> bufA (reused) ----
    {
        const int tilesN = GCN_F_HID / 16;  // 8
        const int blocks = (tilesM * tilesN + (TB / 32) - 1) / (TB / 32);
        gcn_wmma_gemm<1, GCN_F_HID, GCN_F_HID><<<blocks, TB, 0, stream>>>(
            bufA, W2, bufH, tilesM, tilesN);
        const int elems = GCN_N_NODES * GCN_F_HID;
        gcn_agg_init<<<(elems + TB - 1) / TB, TB, 0, stream>>>(
            bufH, dinv, b2, bufA, GCN_N_NODES, GCN_F_HID);
        gcn_edge_scatter<GCN_F_HID><<<nBlkEdgeWaves, TB, 0, stream>>>(
            src, dst, ew, dinv, bufH, bufA, GCN_N_EDGES);
    }

    // ---- layer 3: h3 = relu(agg2) @ W3 ; agg -> d_out (no relu) ----
    {
        const int tilesN = (GCN_F_OUT + 15) / 16;  // 3 (ragged, branch-free guard)
        const int blocks = (tilesM * tilesN + (TB / 32) - 1) / (TB / 32);
        gcn_wmma_gemm<1, GCN_F_HID, GCN_F_OUT><<<blocks, TB, 0, stream>>>(
            bufA, W3, bufH3, tilesM, tilesN);
        const int elems = GCN_N_NODES * GCN_F_OUT;
        gcn_agg_init<<<(elems + TB - 1) / TB, TB, 0, stream>>>(
            bufH3, dinv, b3, out, GCN_N_NODES, GCN_F_OUT);
        gcn_edge_scatter<GCN_F_OUT><<<nBlkEdgeWaves, TB, 0, stream>>>(
            src, dst, ew, dinv, bufH3, out, GCN_N_EDGES);
    }

    (void)in_sizes; (void)n_in; (void)out_size; (void)ws_size;
}